// Generator_24592982737248
// MI455X (gfx1250) — compile-verified
//
#include <hip/hip_runtime.h>

// out[bt,j,n] = sum_{k<64} r[bt,n,k]*cos(2*pi*k*j/1024) - i[bt,n,k]*sin(2*pi*k*j/1024)
// == GEMM D[1024 x 512] = W[1024 x 128] * C[128 x 512] in fp32 via v_wmma_f32_16x16x4_f32.
//   W[j,kk]  = kk<64 ? cos(2*pi*kk*j/1024) : -sin(2*pi*(kk-64)*j/1024)  (LDS table, sin pre-negated)
//   C[kk,cc] = kk<64 ? r[cc,kk] : i[cc,kk-64]   (cc = bt*4+n; x layout [cc][r 128 | i 128])

typedef float v2f __attribute__((ext_vector_type(2)));
typedef float v8f __attribute__((ext_vector_type(8)));

#define TWO_PI_OVER_1024 6.135923151542565e-03f

__global__ __launch_bounds__(256) void ifourier_wmma_f32(const float* __restrict__ x,
                                                         float* __restrict__ out) {
  // tbl[m] = {cos(2*pi*m/1024), -sin(2*pi*m/1024)} — sign folded in so the hot
  // loop has zero negations; all transcendentals hoisted out of the K-loop.
  __shared__ v2f tbl[1024];  // 8 KB of the WGP's 320 KB LDS

  // One-time fill: exact integer quadrant reduction (theta in [-pi/4, pi/4])
  // + minimax polys (~1-2 ulp) — no libm, ~15 VALU per entry, 4 entries/thread.
  for (int m = threadIdx.x; m < 1024; m += 256) {
    const int   n  = (m + 128) >> 8;                 // quadrant 0..4
    const int   r  = m - (n << 8);                   // [-128, 128)
    const float t  = (float)r * TWO_PI_OVER_1024;    // [-pi/4, pi/4]
    const float t2 = t * t;
    // sin(t), |t| <= pi/4 (Cephes minimax, odd deg-7)
    const float s = t * fmaf(t2, fmaf(t2, fmaf(t2, -1.9515295891e-4f,
                                               8.3321608736e-3f),
                                      -1.6666654611e-1f), 1.0f);
    // cos(t), |t| <= pi/4 (Cephes minimax, even deg-8)
    const float c = fmaf(t2, fmaf(t2, fmaf(t2, fmaf(t2, 2.443315711809948e-5f,
                                                    -1.388731625493765e-3f),
                                           4.166664568298827e-2f),
                                  -0.5f), 1.0f);
    // quadrant mapping: angle = n*pi/2 + t
    float cv = (n & 1) ? -s : c;
    float sv = (n & 1) ?  c : s;
    if (n & 2) { cv = -cv; sv = -sv; }
    tbl[m] = (v2f){cv, -sv};                         // store {cos, -sin}
  }
  __syncthreads();

  const int lane = threadIdx.x & 31;
  const int wave = threadIdx.x >> 5;
  const int tile = blockIdx.x * 8 + wave;   // 2048 wave-tiles total
  const int jt   = tile & 63;               // 64 row tiles (j)
  const int ct   = tile >> 6;               // 32 col tiles (bt,n)
  const int rl   = lane & 15;
  const int hi   = lane >> 4;               // 0 -> K{0,1}, 1 -> K{2,3}
  const int j    = jt * 16 + rl;            // A-fragment row for this lane
  const int cc   = ct * 16 + rl;            // B-fragment column for this lane
  const float* __restrict__ xc = x + (size_t)cc * 256;  // [ r[0..127] | i[0..127] ]
  const int kd = hi * 2;

  v8f acc = {};

  // K = 128 fully unrolled: 32 WMMAs, each fed by one ds_load_b64 pair + two
  // global_load_b64 — scheduler pipelines loads across WMMA issue slots.
  #pragma unroll
  for (int s = 0; s < 16; ++s) {
    const int k0 = 4 * s + kd;              // even -> 8B-aligned pair loads
    const int k1 = k0 + 1;

    const v2f cs0 = tbl[(k0 * j) & 1023];   // {cos, -sin}(2*pi*k0*j/1024)
    const v2f cs1 = tbl[(k1 * j) & 1023];

    const v2f a_cos = {cs0.x, cs1.x};       // W rows, cos block (kk = k0,k1)
    const v2f a_sin = {cs0.y, cs1.y};       // W rows, -sin block (kk = 64+k0..)
    const v2f b_cos = *(const v2f*)(xc + k0);        // r[k0], r[k1]
    const v2f b_sin = *(const v2f*)(xc + 128 + k0);  // i[k0], i[k1]

    acc = __builtin_amdgcn_wmma_f32_16x16x4_f32(false, a_cos, false, b_cos,
                                                (short)0, acc, false, false);
    acc = __builtin_amdgcn_wmma_f32_16x16x4_f32(false, a_sin, false, b_sin,
                                                (short)0, acc, false, false);
  }

  // D layout: VGPR g -> M = (hi ? 8+g : g), lane (rl) -> N = cc.
  // Groups of 4 lanes (n = 0..3) write contiguous 16B -> coalesced stores.
  const int bt = cc >> 2;
  const int n  = cc & 3;
  float* __restrict__ op = out + (size_t)bt * 4096 + n;
  const int jbase = jt * 16 + hi * 8;
  #pragma unroll
  for (int g = 0; g < 8; ++g) {
    op[(size_t)(jbase + g) * 4] = acc[g];
  }
}

extern "C" void kernel_launch(void* const* d_in, const int* in_sizes, int n_in,
                              void* d_out, int out_size, void* d_ws, size_t ws_size,
                              hipStream_t stream) {
  (void)in_sizes; (void)n_in; (void)d_ws; (void)ws_size; (void)out_size;
  const float* x = (const float*)d_in[0];   // (8,16,4,2,128) fp32
  float* out = (float*)d_out;               // (8,16,1024,4) fp32
  // 2048 wave-tiles, 8 waves (256 threads) per block -> 256 blocks
  ifourier_wmma_f32<<<256, 256, 0, stream>>>(x, out);
}